// fast_BEV_10488310137170
// MI455X (gfx1250) — compile-verified
//
#include <hip/hip_runtime.h>
#include <hip/hip_bf16.h>

typedef __attribute__((ext_vector_type(2))) float v2f;
typedef __attribute__((ext_vector_type(8))) float v8f;

#define NCAM 6
#define CIN 256
#define FH 32
#define FW 88
#define HWF (FH*FW)          // 2816
#define CAMSTRIDE (CIN*HWF)  // 720896
#define NX 200
#define NY 200
#define NZ 4
#define NPIX (NX*NY)         // 40000
#define COUT 80
// bev tile in LDS: K-pair interleaved float2 layout [kp][px], kp = c/2 (128 groups),
// group pitch 33 float2 (32 px + 1 pad) -> 128*33*8 = 33.8 KB
#define KPITCH 33
#define BEVN (128 * KPITCH)

// ---------------- setup: projection matrices + fused BN affine ----------------
__global__ void setup_kernel(const float* __restrict__ l2i,   // (6,4,4)
                             const float* __restrict__ iam,   // (6,4,4)
                             const float* __restrict__ lam,   // (4,4)
                             const float* __restrict__ convb,
                             const float* __restrict__ gamma,
                             const float* __restrict__ beta,
                             const float* __restrict__ mean,
                             const float* __restrict__ var,
                             float* __restrict__ proj,        // 6*12
                             float* __restrict__ augt,        // 6*2
                             float* __restrict__ scale,       // 80
                             float* __restrict__ shift) {     // 80
  int t = threadIdx.x;
  if (t < NCAM) {
    const float* A = iam + t * 16;
    const float* B = l2i + t * 16;
    float Ar[16], T[16];
    for (int r = 0; r < 4; ++r)
      for (int c = 0; c < 4; ++c)
        Ar[r * 4 + c] = (c == 3) ? 0.0f : A[r * 4 + c];
    for (int r = 0; r < 4; ++r)
      for (int c = 0; c < 4; ++c) {
        float s = 0.f;
        for (int q = 0; q < 4; ++q) s += Ar[r * 4 + q] * B[q * 4 + c];
        T[r * 4 + c] = s;
      }
    for (int r = 0; r < 3; ++r)
      for (int c = 0; c < 4; ++c) {
        float s = 0.f;
        for (int q = 0; q < 4; ++q) s += T[r * 4 + q] * lam[q * 4 + c];
        proj[t * 12 + r * 4 + c] = s;
      }
    augt[t * 2 + 0] = A[0 * 4 + 3];
    augt[t * 2 + 1] = A[1 * 4 + 3];
  }
  if (t >= 16 && t < 16 + COUT) {
    int o = t - 16;
    float s = gamma[o] * rsqrtf(var[o] + 1e-5f);
    scale[o] = s;
    shift[o] = (convb[o] - mean[o]) * s + beta[o];
  }
}

// ---------------- geometry: per-voxel camera selection -> gather offset + weight ----------------
__global__ __launch_bounds__(256) void geom_kernel(const float* __restrict__ pts_w, // (4,200,200)
                                                   const float* __restrict__ proj,
                                                   const float* __restrict__ augt,
                                                   int* __restrict__ base_arr,      // 160000
                                                   float* __restrict__ wgt_arr,     // 160000
                                                   int transposed) {
  int p = blockIdx.x * blockDim.x + threadIdx.x;
  if (p >= NPIX * NZ) return;
  int k = p & 3;          // nz = 4 fastest
  int pj = p >> 2;        // i*200 + j
  int i = pj / NY;
  int j = pj - i * NY;
  float x = (float)i * 0.5f - 50.0f;
  float y = (float)j * 0.5f - 50.0f;
  float z = (float)k * 1.5f - 4.0f;
  int sel = -1, us = 0, vs = 0;
  for (int n = 0; n < NCAM; ++n) {
    const float* P = proj + n * 12;
    float X = P[0] * x + P[1] * y + P[2]  * z + P[3];
    float Y = P[4] * x + P[5] * y + P[6]  * z + P[7];
    float Z = P[8] * x + P[9] * y + P[10] * z + P[11];
    float Zs = (fabsf(Z) > 1e-6f) ? Z : 1e-6f;
    float u = X / Zs + augt[n * 2 + 0];
    float v = Y / Zs + augt[n * 2 + 1];
    int uf = (int)rintf(u * 0.125f);   // stride = 8, round-half-even like jnp.round
    int vf = (int)rintf(v * 0.125f);
    if (uf >= 0 && vf >= 0 && uf < FW && vf < FH && Z > 0.0f) {
      sel = n; us = uf; vs = vf;       // keep LAST valid camera
    }
  }
  bool any = (sel >= 0);
  int r = vs * FW + us;
  int base = 0;
  if (any) base = transposed ? (sel * HWF + r) * CIN : sel * CAMSTRIDE + r;
  base_arr[p] = base;
  wgt_arr[p]  = any ? pts_w[k * NPIX + pj] : 0.0f;
}

// ---------------- (N,C,H,W) -> (N,H,W,C) transpose, LDS-tiled ----------------
__global__ __launch_bounds__(256) void transpose_kernel(const float* __restrict__ in,
                                                        float* __restrict__ out) {
  __shared__ float tile[32][33];
  int cam = blockIdx.z;
  int hw0 = blockIdx.x * 32;     // 2816/32 = 88 tiles
  int c0  = blockIdx.y * 32;     // 256/32  = 8 tiles
  int tx = threadIdx.x;          // 0..31
  int ty = threadIdx.y;          // 0..7
  const float* src = in + cam * CAMSTRIDE;
  float* dst = out + cam * CAMSTRIDE;
#pragma unroll
  for (int i = 0; i < 32; i += 8)
    tile[ty + i][tx] = src[(c0 + ty + i) * HWF + hw0 + tx];
  __syncthreads();
#pragma unroll
  for (int i = 0; i < 32; i += 8)
    dst[(hw0 + ty + i) * CIN + c0 + tx] = tile[tx][ty + i];
}

// Shared phase-2: waves 0-4 each compute one 16-row M-tile for BOTH 16-px N-tiles,
// sharing each conv_w A-fragment between two WMMAs. B operands are single aligned
// 8-byte LDS elements (K-pair interleaved layout) -> one ds_load_b64 each, no shuffles.
__device__ __forceinline__ void wmma_conv_epilogue(
    const float2* __restrict__ bev_s, const float* __restrict__ convw,
    const float* __restrict__ scale, const float* __restrict__ shift,
    float* __restrict__ out, int pix0, int wave, int lane) {
  if (wave >= 5) return;
  int nidx = lane & 15;            // N (and A-row M) index within tile
  int kb   = (lane >> 4) << 1;     // K sub-offset: lanes 0-15 -> {0,1}, 16-31 -> {2,3}
  int m0 = wave * 16;
  const float* arow = convw + (m0 + nidx) * CIN;
  v8f acc0 = {}, acc1 = {};
  for (int k0 = 0; k0 < CIN; k0 += 4) {
    v2f a;
    a.x = arow[k0 + kb];
    a.y = arow[k0 + kb + 1];
    int kp = (k0 + kb) >> 1;       // K-pair group index
    float2 t0 = bev_s[kp * KPITCH + nidx];        // rows (k0+kb, k0+kb+1), col nidx
    float2 t1 = bev_s[kp * KPITCH + nidx + 16];   // same rows, col nidx+16
    v2f b0; b0.x = t0.x; b0.y = t0.y;
    v2f b1; b1.x = t1.x; b1.y = t1.y;
    acc0 = __builtin_amdgcn_wmma_f32_16x16x4_f32(
        false, a, false, b0, (short)0, acc0, false, false);
    acc1 = __builtin_amdgcn_wmma_f32_16x16x4_f32(
        false, a, false, b1, (short)0, acc1, false, false);
  }
  // C/D layout: VGPR r -> M = r (lanes 0-15) or r+8 (lanes 16-31); N = lane%16
  int mofs = (lane >= 16) ? 8 : 0;
#pragma unroll
  for (int r = 0; r < 8; ++r) {
    int o = m0 + r + mofs;
    float s = scale[o], sh = shift[o];
    float v0 = fmaf(acc0[r], s, sh);
    float v1 = fmaf(acc1[r], s, sh);
    v0 = v0 > 0.f ? v0 : 0.f;
    v1 = v1 > 0.f ? v1 : 0.f;
    out[o * NPIX + pix0 + nidx]      = v0;
    out[o * NPIX + pix0 + 16 + nidx] = v1;
  }
}

// ---------------- fused gather (coalesced) + z-reduce + WMMA conv + BN + ReLU ----------------
__global__ __launch_bounds__(256) void bev_conv_coalesced(
    const float* __restrict__ featT,     // (6,32,88,256)
    const int* __restrict__ pos_arr,     // (pix,k) -> (n*2816+v*88+u)*256
    const float* __restrict__ wgt_arr,
    const float* __restrict__ convw,     // (80,256)
    const float* __restrict__ scale,
    const float* __restrict__ shift,
    float* __restrict__ out) {           // (80,200,200)
  __shared__ float2 bev_s[BEVN];         // [kp][px] K-pair interleaved, 33.8 KB
  __shared__ int   pos_s[32 * NZ];
  __shared__ float wgt_s[32 * NZ];

  int tid = threadIdx.x;
  int pix0 = blockIdx.x * 32;
  if (tid < 32 * NZ) {
    pos_s[tid] = pos_arr[pix0 * NZ + tid];
    wgt_s[tid] = wgt_arr[pix0 * NZ + tid];
  }
  __syncthreads();

  int wave = tid >> 5;
  int lane = tid & 31;

  // Phase 1: one wave per pixel; lane l loads channels 8l..8l+7 as contiguous float4 x2
  for (int t = 0; t < 4; ++t) {
    int px = wave + 8 * t;
    float a0 = 0.f, a1 = 0.f, a2 = 0.f, a3 = 0.f;
    float a4 = 0.f, a5 = 0.f, a6 = 0.f, a7 = 0.f;
#pragma unroll
    for (int k = 0; k < NZ; ++k) {
      float w = wgt_s[px * NZ + k];
      const float4* s = (const float4*)(featT + pos_s[px * NZ + k] + lane * 8);
      float4 v0 = s[0];
      float4 v1 = s[1];
      a0 = fmaf(w, v0.x, a0); a1 = fmaf(w, v0.y, a1);
      a2 = fmaf(w, v0.z, a2); a3 = fmaf(w, v0.w, a3);
      a4 = fmaf(w, v1.x, a4); a5 = fmaf(w, v1.y, a5);
      a6 = fmaf(w, v1.z, a6); a7 = fmaf(w, v1.w, a7);
    }
    int kp0 = lane * 4;  // channels 8l..8l+7 = K-pair groups 4l..4l+3
    bev_s[(kp0 + 0) * KPITCH + px] = make_float2(a0, a1);
    bev_s[(kp0 + 1) * KPITCH + px] = make_float2(a2, a3);
    bev_s[(kp0 + 2) * KPITCH + px] = make_float2(a4, a5);
    bev_s[(kp0 + 3) * KPITCH + px] = make_float2(a6, a7);
  }
  __syncthreads();

  wmma_conv_epilogue(bev_s, convw, scale, shift, out, pix0, wave, lane);
}

// ---------------- fallback: scattered gather straight from (N,C,H,W) ----------------
__global__ __launch_bounds__(256) void bev_conv_scatter(
    const float* __restrict__ feat,
    const int* __restrict__ base_arr,
    const float* __restrict__ wgt_arr,
    const float* __restrict__ convw,
    const float* __restrict__ scale,
    const float* __restrict__ shift,
    float* __restrict__ out) {
  __shared__ float2 bev_s[BEVN];
  __shared__ int   base_s[32 * NZ];
  __shared__ float wgt_s[32 * NZ];

  int tid = threadIdx.x;
  int pix0 = blockIdx.x * 32;
  if (tid < 32 * NZ) {
    base_s[tid] = base_arr[pix0 * NZ + tid];
    wgt_s[tid]  = wgt_arr[pix0 * NZ + tid];
  }
  __syncthreads();

  {
    // thread = channel c; write scalar halves of interleaved float2 layout
    float* bevf = (float*)bev_s;
    int c = tid;
    int half = c & 1;
    int kp = c >> 1;
    const float* fc = feat + c * HWF;
    for (int px = 0; px < 32; ++px) {
      float acc = 0.f;
#pragma unroll
      for (int k = 0; k < NZ; ++k)
        acc = fmaf(wgt_s[px * NZ + k], fc[base_s[px * NZ + k]], acc);
      bevf[(kp * KPITCH + px) * 2 + half] = acc;
    }
  }
  __syncthreads();

  wmma_conv_epilogue(bev_s, convw, scale, shift, out, pix0,
                     threadIdx.x >> 5, threadIdx.x & 31);
}

extern "C" void kernel_launch(void* const* d_in, const int* in_sizes, int n_in,
                              void* d_out, int out_size, void* d_ws, size_t ws_size,
                              hipStream_t stream) {
  const float* feat   = (const float*)d_in[0];   // mlvl_feats (1,6,256,32,88)
  const float* points = (const float*)d_in[1];   // (1,4,200,200)
  const float* l2i    = (const float*)d_in[5];   // lidar2image
  const float* iam    = (const float*)d_in[8];   // img_aug_matrix
  const float* lam    = (const float*)d_in[9];   // lidar_aug_matrix
  const float* convw  = (const float*)d_in[11];
  const float* convb  = (const float*)d_in[12];
  const float* gamma  = (const float*)d_in[13];
  const float* beta   = (const float*)d_in[14];
  const float* mean   = (const float*)d_in[15];
  const float* var    = (const float*)d_in[16];

  float* ws   = (float*)d_ws;
  float* proj = ws;                         // 72
  float* augt = ws + 72;                    // 12
  float* scl  = ws + 96;                    // 80
  float* shf  = ws + 176;                   // 80
  int*   base = (int*)(ws + 256);           // 160000
  float* wgt  = ws + 256 + NPIX * NZ;       // 160000
  float* featT = ws + 256 + 2 * NPIX * NZ;  // 4,325,376

  size_t need_full = (size_t)(256 + 2 * NPIX * NZ + NCAM * CAMSTRIDE) * sizeof(float);
  int big = (ws_size >= need_full) ? 1 : 0;

  setup_kernel<<<1, 96, 0, stream>>>(l2i, iam, lam, convb, gamma, beta, mean, var,
                                     proj, augt, scl, shf);
  geom_kernel<<<(NPIX * NZ + 255) / 256, 256, 0, stream>>>(points, proj, augt, base, wgt, big);
  if (big) {
    transpose_kernel<<<dim3(HWF / 32, CIN / 32, NCAM), dim3(32, 8), 0, stream>>>(feat, featT);
    bev_conv_coalesced<<<NPIX / 32, 256, 0, stream>>>(featT, base, wgt, convw, scl, shf,
                                                      (float*)d_out);
  } else {
    bev_conv_scatter<<<NPIX / 32, 256, 0, stream>>>(feat, base, wgt, convw, scl, shf,
                                                    (float*)d_out);
  }
}